// Feedback_8873402434192
// MI455X (gfx1250) — compile-verified
//
#include <hip/hip_runtime.h>

#define UNITS 1024
#define FEAT 128
#define TSEQ 48
#define BATCH 2048
#define OUT_STEPS 24
#define KTOT (FEAT + UNITS)   // 1152
#define KC 32
#define NCHUNK (KTOT / KC)    // 36
#define APAD 36               // A row stride (floats): 16B-aligned float4, conflict-free reads
#define BPAD 40               // B row stride (floats): 16B-aligned float4, conflict-free reads

typedef __attribute__((ext_vector_type(2))) float v2f;
typedef __attribute__((ext_vector_type(8))) float v8f;
typedef __attribute__((ext_vector_type(4))) int   v4i;

#define LDS_AS __attribute__((address_space(3)))
#define GLB_AS __attribute__((address_space(1)))

#if defined(__AMDGCN__) && __has_builtin(__builtin_amdgcn_global_load_async_to_lds_b128)
#define HAVE_ASYNC_LDS 1
#pragma message("CDNA5 probe: global_load_async_to_lds_b128 builtin FOUND - async path enabled")
#else
#define HAVE_ASYNC_LDS 0
#pragma message("CDNA5 probe: global_load_async_to_lds_b128 builtin NOT found - sync LDS staging")
#endif

#if defined(__AMDGCN__) && __has_builtin(__builtin_amdgcn_s_wait_asynccnt)
#define HAVE_WAIT_ASYNC_BUILTIN 1
#else
#define HAVE_WAIT_ASYNC_BUILTIN 0
#endif

__device__ __forceinline__ void copy16_to_lds(const float* g, float* l) {
#if HAVE_ASYNC_LDS
    // global -> LDS DMA, 16B per lane, tracked by ASYNCcnt
    __builtin_amdgcn_global_load_async_to_lds_b128(
        (GLB_AS v4i*)(uintptr_t)g,
        (LDS_AS v4i*)(uint32_t)(uintptr_t)l,
        0, 0);
#else
    *(float4*)l = *(const float4*)g;   // dst 16B-aligned by APAD/BPAD construction
#endif
}

template <int N>
__device__ __forceinline__ void wait_async_lds() {
#if HAVE_ASYNC_LDS
#if HAVE_WAIT_ASYNC_BUILTIN
    __builtin_amdgcn_s_wait_asynccnt((short)N);
#else
    asm volatile("s_wait_asynccnt %0" :: "i"(N) : "memory");
#endif
#endif
}

__device__ __forceinline__ float sigmoidf_(float x) {
    return 1.0f / (1.0f + __expf(-x));
}

// One LSTM time step, fused: z = [x|h] @ [Wk;Wr] + b, gates, c/h update.
// 256 threads = 8 waves (4 M-waves x 2 N-waves); tile 64 rows x 32 units x 4 gates.
// LDS double-buffered; async DMA staging overlapped with WMMA compute.
__global__ __launch_bounds__(256) void lstm_step_kernel(
    const float* __restrict__ x, int x_stride,
    const float* __restrict__ Wk,                  // [FEAT, 4U]
    const float* __restrict__ Wr,                  // [UNITS, 4U]
    const float* __restrict__ bias,                // [4U]
    const float* __restrict__ h_in,                // [B, U]
    float* __restrict__ h_out,                     // [B, U]
    float* cbuf)                                   // [B, U], in place
{
    __shared__ float Alds[2][64][APAD];
    __shared__ float Blds[2][4][KC][BPAD];

    const int tid   = threadIdx.x;
    const int lane  = tid & 31;
    const int wave  = tid >> 5;
    const int mw    = wave >> 1;      // 0..3
    const int nw    = wave & 1;       // 0..1
    const int l15   = lane & 15;
    const int khalf = lane >> 4;      // 0 or 1

    const int Mbase = blockIdx.x * 64;
    const int Nbase = blockIdx.y * 32;

    v8f acc[4] = {};

    // issue one chunk's staging (6 x b128 per thread -> 6 async ops per wave)
    auto issue = [&](int chunk, int bsel) {
        const int kc0 = chunk * KC;
        #pragma unroll
        for (int ii = 0; ii < 2; ++ii) {           // A: 64 x 32 floats
            int idx = tid + ii * 256;
            int r   = idx >> 3;
            int c4  = (idx & 7) << 2;
            int row = Mbase + r;
            int kk  = kc0 + c4;
            const float* src = (kk < FEAT)
                ? (x + (long)row * x_stride + kk)
                : (h_in + (long)row * UNITS + (kk - FEAT));
            copy16_to_lds(src, &Alds[bsel][r][c4]);
        }
        #pragma unroll
        for (int ii = 0; ii < 4; ++ii) {           // B: 4 x 32 x 32 floats
            int idx = tid + ii * 256;
            int c4  = (idx & 7) << 2;
            int kr  = (idx >> 3) & 31;
            int g   = idx >> 8;
            int kk  = kc0 + kr;
            int col = g * UNITS + Nbase + c4;
            const float* src = (kk < FEAT)
                ? (Wk + (long)kk * (4 * UNITS) + col)
                : (Wr + (long)(kk - FEAT) * (4 * UNITS) + col);
            copy16_to_lds(src, &Blds[bsel][g][kr][c4]);
        }
    };

    issue(0, 0);
    for (int c = 0; c < NCHUNK; ++c) {
        const int cur = c & 1;
        if (c + 1 < NCHUNK) {
            issue(c + 1, cur ^ 1);     // prefetch next chunk into other buffer
            wait_async_lds<6>();       // ASYNCcnt retires in order: chunk c has landed
        } else {
            wait_async_lds<0>();
        }
        __syncthreads();               // publish LDS across waves

        #pragma unroll
        for (int k4 = 0; k4 < KC; k4 += 4) {
            v2f a;
            a.x = Alds[cur][mw * 16 + l15][k4 + khalf * 2];
            a.y = Alds[cur][mw * 16 + l15][k4 + khalf * 2 + 1];
            #pragma unroll
            for (int g = 0; g < 4; ++g) {
                v2f b;
                b.x = Blds[cur][g][k4 + khalf * 2][nw * 16 + l15];
                b.y = Blds[cur][g][k4 + khalf * 2 + 1][nw * 16 + l15];
                acc[g] = __builtin_amdgcn_wmma_f32_16x16x4_f32(
                    false, a, false, b, (short)0, acc[g], false, false);
            }
        }
        __syncthreads();               // all waves done reading 'cur' before it is refilled
    }

    // fused gate epilogue: c' = f*c + i*g ; h' = o*tanh(c')
    const int u  = Nbase + nw * 16 + l15;
    const float bi = bias[u];
    const float bf = bias[UNITS + u];
    const float bg = bias[2 * UNITS + u];
    const float bo = bias[3 * UNITS + u];
    #pragma unroll
    for (int j = 0; j < 8; ++j) {
        int row  = Mbase + mw * 16 + khalf * 8 + j;
        long off = (long)row * UNITS + u;
        float ig = sigmoidf_(acc[0][j] + bi);
        float fg = sigmoidf_(acc[1][j] + bf);
        float gg = tanhf(acc[2][j] + bg);
        float og = sigmoidf_(acc[3][j] + bo);
        float cn = fg * cbuf[off] + ig * gg;
        cbuf[off]  = cn;
        h_out[off] = og * tanhf(cn);
    }
}

// y = h @ dense_w + dense_b ; writes feedback buffer and output slice [B, OUT_STEPS, FEAT]
__global__ __launch_bounds__(256) void dense_kernel(
    const float* __restrict__ h,      // [B, UNITS]
    const float* __restrict__ dw,     // [UNITS, FEAT]
    const float* __restrict__ db,     // [FEAT]
    float* __restrict__ ybuf,         // [B, FEAT]
    float* __restrict__ out,          // [B, OUT_STEPS, FEAT]
    int step)
{
    const int tid   = threadIdx.x;
    const int lane  = tid & 31;
    const int wave  = tid >> 5;
    const int mw    = wave >> 1;
    const int nw    = wave & 1;
    const int l15   = lane & 15;
    const int khalf = lane >> 4;
    const int Mbase = blockIdx.x * 64;
    const int Nbase = blockIdx.y * 32;
    const int arow  = Mbase + mw * 16 + l15;
    const int col   = Nbase + nw * 16 + l15;

    v8f acc = {};
    for (int k = 0; k < UNITS; k += 4) {
        v2f a, b;
        const float* ap = h + (long)arow * UNITS + k + khalf * 2;
        a.x = ap[0];
        a.y = ap[1];
        b.x = dw[(long)(k + khalf * 2) * FEAT + col];
        b.y = dw[(long)(k + khalf * 2 + 1) * FEAT + col];
        acc = __builtin_amdgcn_wmma_f32_16x16x4_f32(
            false, a, false, b, (short)0, acc, false, false);
    }
    const float bb = db[col];
    #pragma unroll
    for (int j = 0; j < 8; ++j) {
        int row = Mbase + mw * 16 + khalf * 8 + j;
        float y = acc[j] + bb;
        ybuf[(long)row * FEAT + col] = y;
        out[(long)row * (OUT_STEPS * FEAT) + (long)step * FEAT + col] = y;
    }
}

extern "C" void kernel_launch(void* const* d_in, const int* in_sizes, int n_in,
                              void* d_out, int out_size, void* d_ws, size_t ws_size,
                              hipStream_t stream) {
    (void)in_sizes; (void)n_in; (void)out_size; (void)ws_size;

    const float* inputs = (const float*)d_in[0];  // [B, T, FEAT]
    const float* Wk     = (const float*)d_in[1];  // [FEAT, 4U]
    const float* Wr     = (const float*)d_in[2];  // [UNITS, 4U]
    const float* bias   = (const float*)d_in[3];  // [4U]
    const float* dw     = (const float*)d_in[4];  // [UNITS, FEAT]
    const float* db     = (const float*)d_in[5];  // [FEAT]
    float* out = (float*)d_out;                   // [B, OUT_STEPS, FEAT]

    float* h0 = (float*)d_ws;
    float* h1 = h0 + (size_t)BATCH * UNITS;
    float* cb = h1 + (size_t)BATCH * UNITS;
    float* yb = cb + (size_t)BATCH * UNITS;

    (void)hipMemsetAsync(h0, 0, (size_t)BATCH * UNITS * sizeof(float), stream);
    (void)hipMemsetAsync(cb, 0, (size_t)BATCH * UNITS * sizeof(float), stream);

    dim3 gridL(BATCH / 64, UNITS / 32);  // 32 x 32 WGs
    dim3 gridD(BATCH / 64, FEAT / 32);   // 32 x 4  WGs

    float* hc = h0;
    float* hn = h1;
    for (int t = 0; t < TSEQ; ++t) {
        lstm_step_kernel<<<gridL, 256, 0, stream>>>(
            inputs + (long)t * FEAT, TSEQ * FEAT, Wk, Wr, bias, hc, hn, cb);
        float* tmp = hc; hc = hn; hn = tmp;
    }
    dense_kernel<<<gridD, 256, 0, stream>>>(hc, dw, db, yb, out, 0);
    for (int s = 1; s < OUT_STEPS; ++s) {
        lstm_step_kernel<<<gridL, 256, 0, stream>>>(
            yb, FEAT, Wk, Wr, bias, hc, hn, cb);
        float* tmp = hc; hc = hn; hn = tmp;
        dense_kernel<<<gridD, 256, 0, stream>>>(hc, dw, db, yb, out, s);
    }
}